// AnchorFreeLoss_335007450057
// MI455X (gfx1250) — compile-verified
//
#include <hip/hip_runtime.h>
#include <math.h>

#define B_ 4
#define P_ 32768
#define G_ 128
#define C_ 80
#define INF_ 100000000.0f
#define EPS_ 1e-7f
#define ALPHA_ 0.25f
#define NBX_ 128          // blocks per batch in main kernel (P_/256)
#define NB_ (NBX_ * B_)   // total blocks = 512

typedef float v2f __attribute__((ext_vector_type(2)));
typedef float v8f __attribute__((ext_vector_type(8)));
typedef int   v4i __attribute__((ext_vector_type(4)));
typedef __attribute__((address_space(1))) v4i g_v4i;   // global
typedef __attribute__((address_space(3))) v4i l_v4i;   // LDS

__device__ __forceinline__ float guardf(float x) {
  return __builtin_isfinite(x) ? x : 0.0f;
}

// ---------------------------------------------------------------------------
// Kernel A: fused assignment + focal/IoU/BCE, one thread per (b, p).
// GT boxes are DMA'd into LDS via the CDNA5 async global->LDS path.
// Emits 5 partials per block: class_sum, bbox_sum, conf_sum, pos_cnt, conf_t_sum
// ---------------------------------------------------------------------------
__global__ __launch_bounds__(256) void afl_main(
    const float* __restrict__ y_true,  const float* __restrict__ bbox_true,
    const float* __restrict__ y_pred,  const float* __restrict__ bbox_pred,
    const float* __restrict__ points,  const float* __restrict__ rrange,
    const float* __restrict__ conf_pred, float* __restrict__ part)
{
  __shared__ float s_box[G_][4];
  __shared__ float s_area[G_];
  __shared__ int   s_lbl[G_];
  __shared__ int   s_valid[G_];
  __shared__ float sred[256];

  const int tid = threadIdx.x;
  const int b   = blockIdx.y;
  const int p   = blockIdx.x * 256 + tid;

  // --- async DMA of the GT box table for this batch into LDS (CDNA5 path) ---
  if (tid < G_) {
    const float* gsrc = bbox_true + ((size_t)b * G_ + tid) * 4;
    __builtin_amdgcn_global_load_async_to_lds_b128(
        (g_v4i*)const_cast<float*>(gsrc),
        (l_v4i*)&s_box[tid][0],
        0, 0);
  }
  // label index per GT from the one-hot rows (tiny; L2-resident)
  for (int g = tid; g < G_; g += 256) {
    const float* row = y_true + ((size_t)b * G_ + g) * C_;
    int lbl = 0;
    for (int c = 0; c < C_; ++c) { if (row[c] > 0.5f) { lbl = c; break; } }
    s_lbl[g] = lbl;
  }
  asm volatile("s_wait_asynccnt 0x0" ::: "memory");
  __syncthreads();
  if (tid < G_) {
    float x1 = s_box[tid][0], y1v = s_box[tid][1];
    float x2 = s_box[tid][2], y2v = s_box[tid][3];
    s_valid[tid] = (x1 > 0.f) || (y1v > 0.f) || (x2 > 0.f) || (y2v > 0.f);
    s_area[tid]  = (x2 - x1) * (y2v - y1v);
  }
  __syncthreads();

  const float px  = points[p * 2 + 0];
  const float py  = points[p * 2 + 1];
  const float rlo = rrange[p * 2 + 0];
  const float rhi = rrange[p * 2 + 1];

  // --- assignment: argmin of area over candidate GTs (first-occurrence) ---
  float best = INF_;
  int   bi   = 0;
  #pragma unroll 4
  for (int g = 0; g < G_; ++g) {
    float x1 = s_box[g][0], y1v = s_box[g][1];
    float x2 = s_box[g][2], y2v = s_box[g][3];
    float l = px - x1, t = py - y1v, r = x2 - px, btm = y2v - py;
    float mino = fminf(fminf(l, t), fminf(r, btm));
    float maxo = fmaxf(fmaxf(l, t), fmaxf(r, btm));
    bool cand = (mino > 0.f) & (rlo <= maxo) & (maxo <= rhi) & (s_valid[g] != 0);
    float cost = cand ? s_area[g] : INF_;
    if (cost < best) { best = cost; bi = g; }
  }
  const bool pos = best < INF_;

  // --- centerness (conf target) ---
  float conf_t = 0.f;
  float bt0 = 0.f, bt1 = 0.f, bt2 = 0.f, bt3 = 0.f;
  if (pos) {
    bt0 = s_box[bi][0]; bt1 = s_box[bi][1];
    bt2 = s_box[bi][2]; bt3 = s_box[bi][3];
    float l = px - bt0, t = py - bt1, r = bt2 - px, btm = bt3 - py;
    float q = (fminf(l, r)   / fmaxf(fmaxf(l, r),   EPS_)) *
              (fminf(t, btm) / fmaxf(fmaxf(t, btm), EPS_));
    conf_t = sqrtf(fmaxf(q, 0.f));
  }

  // --- bbox IoU loss (weighted by conf_t) ---
  const size_t pb = (size_t)b * P_ + p;
  float bp0 = bbox_pred[pb * 4 + 0], bp1 = bbox_pred[pb * 4 + 1];
  float bp2 = bbox_pred[pb * 4 + 2], bp3 = bbox_pred[pb * 4 + 3];
  float ix1 = fmaxf(bt0, bp0), iy1 = fmaxf(bt1, bp1);
  float ix2 = fminf(bt2, bp2), iy2 = fminf(bt3, bp3);
  float inter = fmaxf(ix2 - ix1, 0.f) * fmaxf(iy2 - iy1, 0.f);
  float at = fmaxf(bt2 - bt0, 0.f) * fmaxf(bt3 - bt1, 0.f);
  float ap = fmaxf(bp2 - bp0, 0.f) * fmaxf(bp3 - bp1, 0.f);
  float iou = inter / fmaxf(at + ap - inter, EPS_);
  float bbox_elem = (1.f - iou) * conf_t;

  // --- confidence BCE (weighted by pos) ---
  float cp = fminf(fmaxf(conf_pred[pb], EPS_), 1.f - EPS_);
  float bce = -(conf_t * logf(cp) + (1.f - conf_t) * logf(1.f - cp));
  float conf_elem = pos ? bce : 0.f;

  // --- focal class loss: negative term for all C, single positive correction
  const float* yp  = y_pred + pb * C_;
  const float4* yp4 = (const float4*)yp;
  float acc = 0.f;
  #pragma unroll 5
  for (int i = 0; i < C_ / 4; ++i) {
    float4 v = yp4[i];
    float pr;
    pr = fminf(fmaxf(v.x, EPS_), 1.f - EPS_); acc += pr * pr * logf(1.f - pr);
    pr = fminf(fmaxf(v.y, EPS_), 1.f - EPS_); acc += pr * pr * logf(1.f - pr);
    pr = fminf(fmaxf(v.z, EPS_), 1.f - EPS_); acc += pr * pr * logf(1.f - pr);
    pr = fminf(fmaxf(v.w, EPS_), 1.f - EPS_); acc += pr * pr * logf(1.f - pr);
  }
  acc *= -(1.f - ALPHA_);
  if (pos) {
    int ct = s_lbl[bi];
    float pr = fminf(fmaxf(yp[ct], EPS_), 1.f - EPS_);
    acc += (1.f - ALPHA_) * pr * pr * logf(1.f - pr)
         - ALPHA_ * (1.f - pr) * (1.f - pr) * logf(pr);
  }

  // --- deterministic fixed-tree block reduction of 5 partials ---
  auto reduce = [&](float v) -> float {
    __syncthreads();
    sred[tid] = v;
    __syncthreads();
    #pragma unroll
    for (int s = 128; s > 0; s >>= 1) {
      if (tid < s) sred[tid] += sred[tid + s];
      __syncthreads();
    }
    return sred[0];
  };
  float cls_s  = reduce(acc);
  float bbox_s = reduce(bbox_elem);
  float conf_s = reduce(conf_elem);
  float pos_s  = reduce(pos ? 1.f : 0.f);
  float ct_s   = reduce(conf_t);

  if (tid == 0) {
    int blk = blockIdx.y * NBX_ + blockIdx.x;   // grouped by batch
    part[0 * NB_ + blk] = cls_s;
    part[1 * NB_ + blk] = bbox_s;
    part[2 * NB_ + blk] = conf_s;
    part[3 * NB_ + blk] = pos_s;
    part[4 * NB_ + blk] = ct_s;
  }
}

// ---------------------------------------------------------------------------
// Kernel B: one wave. WMMA ones-matmul reduction of the 512 block partials
// (D = A x ones + C  =>  row sums; total = sum of D column 0). Deterministic.
// ---------------------------------------------------------------------------
__global__ __launch_bounds__(32) void afl_final(const float* __restrict__ part,
                                                float* __restrict__ out)
{
  const int lane = threadIdx.x;
  v2f ones; ones[0] = 1.f; ones[1] = 1.f;
  v8f acc0 = {}; v8f acc1 = {}; v8f acc2 = {};

  #pragma unroll
  for (int i = 0; i < NB_ / 64; ++i) {
    v2f a;
    a[0] = part[0 * NB_ + i * 64 + lane];
    a[1] = part[0 * NB_ + i * 64 + 32 + lane];
    acc0 = __builtin_amdgcn_wmma_f32_16x16x4_f32(false, a, false, ones,
                                                 (short)0, acc0, false, false);
    a[0] = part[1 * NB_ + i * 64 + lane];
    a[1] = part[1 * NB_ + i * 64 + 32 + lane];
    acc1 = __builtin_amdgcn_wmma_f32_16x16x4_f32(false, a, false, ones,
                                                 (short)0, acc1, false, false);
    a[0] = part[2 * NB_ + i * 64 + lane];
    a[1] = part[2 * NB_ + i * 64 + 32 + lane];
    acc2 = __builtin_amdgcn_wmma_f32_16x16x4_f32(false, a, false, ones,
                                                 (short)0, acc2, false, false);
  }

  // total = sum over M of rowsum: rows 0-7 live in lane 0, rows 8-15 in lane 16
  float tcls = 0.f, tbbox = 0.f, tconf = 0.f;
  #pragma unroll
  for (int r = 0; r < 8; ++r) {
    tcls  += __shfl(acc0[r], 0) + __shfl(acc0[r], 16);
    tbbox += __shfl(acc1[r], 0) + __shfl(acc1[r], 16);
    tconf += __shfl(acc2[r], 0) + __shfl(acc2[r], 16);
  }

  if (lane == 0) {
    float avg = 0.f, bavg = 0.f;
    for (int bb = 0; bb < B_; ++bb) {
      float pc = 0.f, cs = 0.f;
      for (int k = 0; k < NBX_; ++k) {
        pc += part[3 * NB_ + bb * NBX_ + k];
        cs += part[4 * NB_ + bb * NBX_ + k];
      }
      avg  += fmaxf(pc, 1.0f);
      bavg += fmaxf(cs, 1e-6f);
    }
    out[0] = guardf(tcls  / avg);
    out[1] = guardf(tbbox / bavg);
    out[2] = guardf(tconf / avg);
  }
}

extern "C" void kernel_launch(void* const* d_in, const int* in_sizes, int n_in,
                              void* d_out, int out_size, void* d_ws, size_t ws_size,
                              hipStream_t stream) {
  const float* y_true    = (const float*)d_in[0];
  const float* bbox_true = (const float*)d_in[1];
  const float* y_pred    = (const float*)d_in[2];
  const float* bbox_pred = (const float*)d_in[3];
  const float* points    = (const float*)d_in[4];
  const float* rrange    = (const float*)d_in[5];
  const float* conf_pred = (const float*)d_in[6];
  float* out  = (float*)d_out;
  float* part = (float*)d_ws;     // 5 * NB_ floats = 10 KB of scratch

  dim3 grid(NBX_, B_);
  afl_main<<<grid, 256, 0, stream>>>(y_true, bbox_true, y_pred, bbox_pred,
                                     points, rrange, conf_pred, part);
  afl_final<<<1, 32, 0, stream>>>(part, out);
}